// CombinedCriterionAE_58085137711543
// MI455X (gfx1250) — compile-verified
//
#include <hip/hip_runtime.h>
#include <math.h>

typedef __attribute__((ext_vector_type(2))) float v2f;
typedef __attribute__((ext_vector_type(8))) float v8f;

#define N_PRED 16384
#define N_GT   32768
#define BETA   0.45f
#define GAMMA  0.45f
#define CHUNK  2048   // gt points staged per LDS pass: 2048 * 16B = 32 KB

// ---------------------------------------------------------------------------
// Prep: pack WMMA operands.
//   A row (per pred):  (-2x, -2y, -2z, 1)
//   B col (per gt):    ( gx,  gy,  gz, |g|^2 )
// => WMMA f32 16x16x4 tile[m][n] = |g_n|^2 - 2 p_m . g_n = d2[m][n] - |p_m|^2
// (|p|^2 is constant per row -> argmin unchanged). Also zero the accumulator.
// ---------------------------------------------------------------------------
__global__ __launch_bounds__(256) void prep_kernel(const float* __restrict__ pred,
                                                   const float* __restrict__ gt,
                                                   float4* __restrict__ pr4,
                                                   float4* __restrict__ gt4,
                                                   float* __restrict__ acc) {
  int tid = blockIdx.x * 256 + threadIdx.x;
  if (tid == 0) acc[0] = 0.0f;
  if (tid < N_PRED) {
    float x = pred[tid * 6 + 0], y = pred[tid * 6 + 1], z = pred[tid * 6 + 2];
    pr4[tid] = make_float4(-2.0f * x, -2.0f * y, -2.0f * z, 1.0f);
  }
  if (tid < N_GT) {
    float x = gt[tid * 6 + 0], y = gt[tid * 6 + 1], z = gt[tid * 6 + 2];
    gt4[tid] = make_float4(x, y, z, x * x + y * y + z * z);
  }
}

// ---------------------------------------------------------------------------
// Main: each wave owns 16 pred rows (one M tile), scans all 32768 gt points in
// LDS-staged chunks, one v_wmma_f32_16x16x4_f32 per 16x16 tile, per-lane
// running argmin, then cross-lane (16-lane half-wave) argmin reduction.
// 128 blocks * 8 waves * 16 rows = 16384 rows, no divergence (EXEC all ones).
// ---------------------------------------------------------------------------
__global__ __launch_bounds__(256) void argmin_kernel(const float4* __restrict__ pr4,
                                                     const float4* __restrict__ gt4,
                                                     int* __restrict__ out_idx) {
  __shared__ float4 sh[CHUNK];

  const int lane = threadIdx.x & 31;
  const int wave = threadIdx.x >> 5;
  const int m0   = (blockIdx.x * 8 + wave) * 16;

  const int nsel = lane & 15;        // this lane's N (and M for A-tile) within the tile
  const int hsel = (lane >> 4) << 1; // float offset: lanes 0-15 -> K0,K1; lanes 16-31 -> K2,K3

  // A tile: lane L<16 holds A[M=L][K=0..1]; lane L>=16 holds A[M=L-16][K=2..3]
  const float* abase = reinterpret_cast<const float*>(&pr4[m0 + nsel]);
  const float2 af = *reinterpret_cast<const float2*>(abase + hsel);
  const v2f a = {af.x, af.y};

  float minv[8];
  int   mini[8];
#pragma unroll
  for (int i = 0; i < 8; ++i) { minv[i] = 3.4e38f; mini[i] = 0; }

  for (int c0 = 0; c0 < N_GT; c0 += CHUNK) {
    __syncthreads();
    // Cooperative stage of CHUNK packed gt points into LDS (b128 per thread).
    for (int t = threadIdx.x; t < CHUNK; t += 256) sh[t] = gt4[c0 + t];
    __syncthreads();

#pragma unroll 2
    for (int n = 0; n < CHUNK; n += 16) {
      // B tile: lane L<16 holds B[K=0..1][N=L]; lane L>=16 holds B[K=2..3][N=L-16]
      const float* bbase = reinterpret_cast<const float*>(&sh[n + nsel]);
      const float2 bf = *reinterpret_cast<const float2*>(bbase + hsel);
      const v2f b = {bf.x, bf.y};

      v8f c = {};
      c = __builtin_amdgcn_wmma_f32_16x16x4_f32(
          /*neg_a=*/false, a, /*neg_b=*/false, b,
          /*c_mod=*/(short)0, c, /*reuse_a=*/false, /*reuse_b=*/false);

      const int gidx = c0 + n + nsel;
#pragma unroll
      for (int i = 0; i < 8; ++i) {
        if (c[i] < minv[i]) { minv[i] = c[i]; mini[i] = gidx; }  // strict < => first index wins
      }
    }
  }

  // C layout: VGPR i = row m0+i (lanes 0-15) and row m0+8+i (lanes 16-31).
  // Reduce argmin across the 16 lanes of each half-wave; tie-break on smaller index.
#pragma unroll
  for (int i = 0; i < 8; ++i) {
    float v  = minv[i];
    int   id = mini[i];
#pragma unroll
    for (int m = 1; m < 16; m <<= 1) {
      float ov  = __shfl_xor(v, m, 32);
      int   oid = __shfl_xor(id, m, 32);
      if (ov < v || (ov == v && oid < id)) { v = ov; id = oid; }
    }
    if ((lane & 15) == 0) out_idx[m0 + ((lane >> 4) << 3) + i] = id;
  }
}

// ---------------------------------------------------------------------------
// Normal-cosine loss: gather gt normal at argmin, normalize both, sum (1-cos).
// ---------------------------------------------------------------------------
__global__ __launch_bounds__(256) void normloss_kernel(const float* __restrict__ pred,
                                                       const float* __restrict__ gt,
                                                       const int* __restrict__ idx,
                                                       float* __restrict__ acc) {
  __shared__ float red[8];
  const int tid = blockIdx.x * 256 + threadIdx.x;

  float px = pred[tid * 6 + 3], py = pred[tid * 6 + 4], pz = pred[tid * 6 + 5];
  const int j = idx[tid];
  float gx = gt[j * 6 + 3], gy = gt[j * 6 + 4], gz = gt[j * 6 + 5];

  float pl = fmaxf(sqrtf(px * px + py * py + pz * pz), 1e-12f);
  float gl = fmaxf(sqrtf(gx * gx + gy * gy + gz * gz), 1e-12f);
  float cosv = (px * gx + py * gy + pz * gz) / (pl * gl);
  float v = 1.0f - cosv;

  // wave32 reduction, then one shared slot per wave, then one atomic per block
#pragma unroll
  for (int m = 16; m >= 1; m >>= 1) v += __shfl_xor(v, m, 32);
  if ((threadIdx.x & 31) == 0) red[threadIdx.x >> 5] = v;
  __syncthreads();
  if (threadIdx.x == 0) {
    float s = 0.0f;
#pragma unroll
    for (int w = 0; w < 8; ++w) s += red[w];
    atomicAdd(acc, s);
  }
}

// ---------------------------------------------------------------------------
// Finalize: scalar regularizer + blend (single thread).
// ---------------------------------------------------------------------------
__global__ void finalize_kernel(const float* __restrict__ R,
                                const float* __restrict__ t,
                                const float* __restrict__ s,
                                const float* __restrict__ acc,
                                float* __restrict__ out) {
  float rl = 0.0f;
#pragma unroll
  for (int i = 0; i < 3; ++i)
#pragma unroll
    for (int jj = 0; jj < 3; ++jj) {
      float d = R[i * 3 + jj] - ((i == jj) ? 1.0f : 0.0f);
      rl += d * d;
    }
  rl = sqrtf(rl);
  float tl = sqrtf(t[0] * t[0] + t[1] * t[1] + t[2] * t[2]);
  float sl = (s[0] - 1.0f) * (s[0] - 1.0f);
  float reg = rl + tl + sl;
  float nloss = acc[0] / (float)N_PRED;
  out[0] = BETA * reg + GAMMA * nloss;
}

extern "C" void kernel_launch(void* const* d_in, const int* in_sizes, int n_in,
                              void* d_out, int out_size, void* d_ws, size_t ws_size,
                              hipStream_t stream) {
  (void)in_sizes; (void)n_in; (void)out_size; (void)ws_size;
  const float* pred = (const float*)d_in[0];  // 16384 x 6
  const float* gt   = (const float*)d_in[1];  // 32768 x 6
  const float* R    = (const float*)d_in[2];  // 3 x 3
  const float* t    = (const float*)d_in[3];  // 3
  const float* s    = (const float*)d_in[4];  // 1
  float* out = (float*)d_out;

  char* ws = (char*)d_ws;
  float4* pr4 = (float4*)(ws);                                   // 256 KB
  float4* gt4 = (float4*)(ws + (size_t)N_PRED * 16);             // 512 KB
  int*    idx = (int*)   (ws + (size_t)N_PRED * 16 + (size_t)N_GT * 16); // 64 KB
  float*  acc = (float*) (ws + (size_t)N_PRED * 16 + (size_t)N_GT * 16 + (size_t)N_PRED * 4);

  prep_kernel<<<N_GT / 256, 256, 0, stream>>>(pred, gt, pr4, gt4, acc);
  argmin_kernel<<<N_PRED / (8 * 16), 256, 0, stream>>>(pr4, gt4, idx);   // 128 blocks
  normloss_kernel<<<N_PRED / 256, 256, 0, stream>>>(pred, gt, idx, acc);
  finalize_kernel<<<1, 1, 0, stream>>>(R, t, s, acc, out);
}